// CreatorXSimGCL_7988639170795
// MI455X (gfx1250) — compile-verified
//
#include <hip/hip_runtime.h>

#define NUM_USERS 50000
#define NUM_ITEMS 100000
#define NTOT      (NUM_USERS + NUM_ITEMS)   // 150000 (multiple of 16)
#define DIM       128
#define N_LAYERS  3
#define EPS_C     0.2f

typedef __attribute__((ext_vector_type(2))) float v2f;
typedef __attribute__((ext_vector_type(8))) float v8f;

// ---------------------------------------------------------------------------
// SpMM via L2-resident atomic scatter: one wave32 per edge.
// Lane L handles dims [4L, 4L+3]: one coalesced 512B gather of x[col],
// four contiguous global_atomic_add_f32 into y[row].
// ---------------------------------------------------------------------------
__global__ void __launch_bounds__(256) spmm_atomic_kernel(
    const int* __restrict__ rows, const int* __restrict__ cols,
    const float* __restrict__ vals, const float* __restrict__ x,
    float* __restrict__ y, int nnz)
{
    const int edge = (int)((blockIdx.x * blockDim.x + threadIdx.x) >> 5);
    const int lane = threadIdx.x & 31;
    if (edge >= nnz) return;

    const int   r = rows[edge];
    const int   c = cols[edge];
    const float v = vals[edge];

    const float4* xr = (const float4*)(x + (size_t)c * DIM);
    const float4  p  = xr[lane];

    float* yr = y + (size_t)r * DIM + (size_t)lane * 4;
    __hip_atomic_fetch_add(yr + 0, v * p.x, __ATOMIC_RELAXED, __HIP_MEMORY_SCOPE_AGENT);
    __hip_atomic_fetch_add(yr + 1, v * p.y, __ATOMIC_RELAXED, __HIP_MEMORY_SCOPE_AGENT);
    __hip_atomic_fetch_add(yr + 2, v * p.z, __ATOMIC_RELAXED, __HIP_MEMORY_SCOPE_AGENT);
    __hip_atomic_fetch_add(yr + 3, v * p.w, __ATOMIC_RELAXED, __HIP_MEMORY_SCOPE_AGENT);
}

// ---------------------------------------------------------------------------
// Post-processing: one wave32 per 16-row tile.
// Sum-of-squares of 16 noise rows via V_WMMA_F32_16X16X4_F32 with B = ones.
// Since B is all-ones, the assignment of dims to WMMA K-slots is arbitrary:
// lane L (row = L%16, khalf = L/16) loads float4 chunks at dim 8*i + 4*khalf,
// covering all 128 dims exactly once across i = 0..15 and both halves.
// Each b128 load feeds TWO WMMAs ((x,y) then (z,w)). All 16 loads are issued
// before the WMMA chain so memory latency overlaps the serial C accumulation.
// C/D layout: lane holds VGPR r = sum for row (r + 8*khalf); shfl_xor(16)
// swaps halves so every lane gets all 16 row norms.
// ---------------------------------------------------------------------------
__global__ void __launch_bounds__(256) post_noise_kernel(
    const float* __restrict__ s,      // SpMM output for this layer
    const float* __restrict__ nz,     // noise[k], N x 128
    float* __restrict__ ego,          // next ego (ping buffer)
    float* __restrict__ acc,          // running average accumulator (d_out)
    int ntiles)
{
    const int tile = (int)((blockIdx.x * blockDim.x + threadIdx.x) >> 5);
    if (tile >= ntiles) return;                 // whole wave exits: EXEC all-1s for WMMA
    const int lane  = threadIdx.x & 31;
    const int m     = lane & 15;
    const int khalf = lane >> 4;                // selects K-slot pair within each A tile

    // Batch-load this lane's 64 noise values as 16 float4s (registers).
    const float* nrow = nz + ((size_t)(tile * 16 + m)) * DIM + khalf * 4;
    float4 q[16];
#pragma unroll
    for (int i = 0; i < 16; i++) {
        q[i] = *(const float4*)(nrow + 8 * i);
    }

    // 32-step WMMA chain: D = A*1 + C accumulates per-row sum of squares.
    v8f cacc = {};
    const v2f ones = {1.0f, 1.0f};
#pragma unroll
    for (int i = 0; i < 16; i++) {
        v2f a;
        a.x = q[i].x * q[i].x;
        a.y = q[i].y * q[i].y;
        cacc = __builtin_amdgcn_wmma_f32_16x16x4_f32(
            false, a, false, ones, (short)0, cacc, false, false);
        a.x = q[i].z * q[i].z;
        a.y = q[i].w * q[i].w;
        cacc = __builtin_amdgcn_wmma_f32_16x16x4_f32(
            false, a, false, ones, (short)0, cacc, false, false);
    }

    // Every lane holds 8 row-sums (rows r + 8*khalf). Swap halves via shfl.
    float invn[16];
#pragma unroll
    for (int r = 0; r < 8; r++) {
        const float mine  = cacc[r];
        const float other = __shfl_xor(mine, 16, 32);
        const float lo = khalf ? other : mine;   // row r
        const float hi = khalf ? mine  : other;  // row r + 8
        invn[r]     = rsqrtf(lo) * EPS_C;
        invn[r + 8] = rsqrtf(hi) * EPS_C;
    }

    // Apply noise injection + accumulate average. Lane covers dims [4L,4L+3].
    const size_t base = (size_t)tile * 16 * DIM + (size_t)lane * 4;
#pragma unroll
    for (int rr = 0; rr < 16; rr++) {
        const size_t off = base + (size_t)rr * DIM;
        const float4 sv = *(const float4*)(s  + off);
        const float4 nv = *(const float4*)(nz + off);
        const float  k  = invn[rr];
        float4 e;
        e.x = sv.x + copysignf(nv.x * k, sv.x);
        e.y = sv.y + copysignf(nv.y * k, sv.y);
        e.z = sv.z + copysignf(nv.z * k, sv.z);
        e.w = sv.w + copysignf(nv.w * k, sv.w);
        *(float4*)(ego + off) = e;

        float4 o = *(float4*)(acc + off);
        o.x += e.x * (1.0f / (float)N_LAYERS);
        o.y += e.y * (1.0f / (float)N_LAYERS);
        o.z += e.z * (1.0f / (float)N_LAYERS);
        o.w += e.w * (1.0f / (float)N_LAYERS);
        *(float4*)(acc + off) = o;
    }
}

// ---------------------------------------------------------------------------
// Host orchestration (graph-capture safe: async memcpy/memset + kernels only)
// ---------------------------------------------------------------------------
extern "C" void kernel_launch(void* const* d_in, const int* in_sizes, int n_in,
                              void* d_out, int out_size, void* d_ws, size_t ws_size,
                              hipStream_t stream)
{
    const float* user_emb = (const float*)d_in[0];
    const float* item_emb = (const float*)d_in[1];
    const float* vals     = (const float*)d_in[2];
    const float* noise    = (const float*)d_in[3];
    const int*   rows     = (const int*)d_in[4];
    const int*   cols     = (const int*)d_in[5];
    float*       out      = (float*)d_out;

    const size_t nd  = (size_t)NTOT * DIM;
    float* cur = (float*)d_ws;        // current ego
    float* tmp = cur + nd;            // SpMM scratch

    const int nnz = in_sizes[2];

    // ego0 = concat(user_emb, item_emb); out = 0
    hipMemcpyAsync(cur, user_emb, sizeof(float) * (size_t)NUM_USERS * DIM,
                   hipMemcpyDeviceToDevice, stream);
    hipMemcpyAsync(cur + (size_t)NUM_USERS * DIM, item_emb,
                   sizeof(float) * (size_t)NUM_ITEMS * DIM,
                   hipMemcpyDeviceToDevice, stream);
    hipMemsetAsync(out, 0, sizeof(float) * nd, stream);

    const int spmm_blocks = (nnz + 7) / 8;          // 8 edges (waves) / 256-thr block
    const int ntiles      = NTOT / 16;              // 9375
    const int post_blocks = (ntiles + 7) / 8;       // 8 tiles (waves) / block

    for (int k = 0; k < N_LAYERS; k++) {
        hipMemsetAsync(tmp, 0, sizeof(float) * nd, stream);
        spmm_atomic_kernel<<<spmm_blocks, 256, 0, stream>>>(rows, cols, vals, cur, tmp, nnz);
        post_noise_kernel<<<post_blocks, 256, 0, stream>>>(
            tmp, noise + (size_t)k * nd, cur, out, ntiles);
    }
}